// MultiGranularitySpaceChaos_40398462386445
// MI455X (gfx1250) — compile-verified
//
#include <hip/hip_runtime.h>

// Pure block-permutation of [B,C,H,W] f32: out block (p/4, p%4) = in block (k/4, k%4)
// with p = perm[b*16 + k].  Zero FLOPs -> HBM-bandwidth-bound (822 MB total,
// ~35us floor @ 23.3 TB/s).  Strategy: B128 nontemporal loads/stores (CDNA5 TH=NT
// cache hints), scatter form so perm lookup is a uniform scalar load per workgroup,
// and a channel-strided unroll so the (c,y,xv) index decomposition is computed once.

typedef float v4f __attribute__((ext_vector_type(4)));

namespace {
constexpr int B_  = 16;
constexpr int C_  = 128;
constexpr int H_  = 224;
constexpr int W_  = 224;
constexpr int G_  = 4;
constexpr int BH_ = H_ / G_;          // 56 rows per block
constexpr int BW_ = W_ / G_;          // 56 cols per block
constexpr int WV   = W_ / 4;          // 56  float4 per full image row
constexpr int SEGV = BW_ / 4;         // 14  float4 per block-row segment (224 B, contiguous)
constexpr int BLKV = BH_ * SEGV;      // 784 float4 per (block, channel)
constexpr int BKV  = C_ * BLKV;       // 100352 float4 per (batch, block)
constexpr int UNROLL  = 4;
constexpr int THREADS = 256;
constexpr int TPBK    = BKV / UNROLL;     // 25088 threads per (b, k)
constexpr int GRIDX   = TPBK / THREADS;   // 98 workgroups in x
constexpr int CSPAN   = TPBK / BLKV;      // 32 channels covered per unroll step
constexpr int STEPV4  = CSPAN * H_ * WV;  // float4 stride between unroll steps
static_assert(TPBK * UNROLL == BKV, "exact tiling");
static_assert(GRIDX * THREADS == TPBK, "exact tiling");
static_assert(CSPAN * BLKV == TPBK, "unroll step is a whole number of channels");
static_assert(CSPAN * UNROLL == C_, "unroll covers all channels");
}

__global__ __launch_bounds__(THREADS)
void MultiGranularitySpaceChaos_block_shuffle(const v4f* __restrict__ x,
                                              const int* __restrict__ perm,
                                              v4f* __restrict__ out) {
    const int b = blockIdx.z;          // batch
    const int k = blockIdx.y;          // source block index 0..15
    // Uniform across the workgroup -> scalar load; pin value to SGPR.
    const int p  = __builtin_amdgcn_readfirstlane(perm[b * (G_ * G_) + k]);
    const int si = k >> 2, sj = k & 3; // source block coords
    const int di = p >> 2, dj = p & 3; // destination block coords

    // Decompose ONCE: t -> (c0, y, xv).  Unroll steps advance the channel by
    // exactly CSPAN (=32), i.e. a constant float4 offset STEPV4 -- no per-step
    // division chains, identical fully-coalesced access pattern.
    const int t  = blockIdx.x * THREADS + threadIdx.x;   // [0, TPBK)
    const int c0 = t / BLKV;                             // [0, CSPAN)
    const int r  = t - c0 * BLKV;                        // [0, 784)
    const int y  = r / SEGV;                             // row within block, [0,56)
    const int xv = r - y * SEGV;                         // float4 within segment, [0,14)

    const int plane = (b * C_ + c0) * H_;                // row index of (b,c0,row=0)
    const int src0 = (plane + si * BH_ + y) * WV + sj * SEGV + xv;
    const int dst0 = (plane + di * BH_ + y) * WV + dj * SEGV + xv;

#pragma unroll
    for (int u = 0; u < UNROLL; ++u) {
        // Single-touch streaming: NT temporal hint keeps 822 MB of traffic
        // from thrashing the 192 MB L2 (gfx1250 TH field on B128 vmem ops).
        v4f v = __builtin_nontemporal_load(x + src0 + u * STEPV4);
        __builtin_nontemporal_store(v, out + dst0 + u * STEPV4);
    }
}

extern "C" void kernel_launch(void* const* d_in, const int* in_sizes, int n_in,
                              void* d_out, int out_size, void* d_ws, size_t ws_size,
                              hipStream_t stream) {
    (void)in_sizes; (void)n_in; (void)out_size; (void)d_ws; (void)ws_size;
    const v4f* x    = (const v4f*)d_in[0];   // [16,128,224,224] f32
    const int* perm = (const int*)d_in[1];   // [16,16] i32
    v4f* out        = (v4f*)d_out;           // [16,128,224,224] f32

    dim3 grid(GRIDX, G_ * G_, B_);           // (98, 16, 16)
    MultiGranularitySpaceChaos_block_shuffle<<<grid, THREADS, 0, stream>>>(x, perm, out);
}